// Model_33767032881834
// MI455X (gfx1250) — compile-verified
//
#include <hip/hip_runtime.h>
#include <cstddef>

// ---------------------------------------------------------------------------
// 3-layer LSTM (B=256,S=512,H=512) + FC head for MI455X (gfx1250, wave32).
// Matrix work uses V_WMMA_F32_16X16X4_F32 (fp32 in/out -> reference numerics).
// A-tile staging uses CDNA5 async LDS loads (GLOBAL_LOAD_ASYNC_TO_LDS_B128,
// ASYNCcnt) with double-buffered K-chunks.
//
// Workspace layout (floats), ~1.64 GB total:
//   WhhT0..2, WihT1..2 : 5 x 512*2048      (transposed weights, L2-resident)
//   c, c0, z           : 3 x 256*512       (cell state, layer-0 final c, zeros)
//   out0, out1         : 2 x 256*512*512   (h sequences; layer 2 reuses out0)
//   xp                 : 256*512*2048      (precomputed input projections)
// ---------------------------------------------------------------------------

typedef __attribute__((ext_vector_type(2))) float v2f;
typedef __attribute__((ext_vector_type(8))) float v8f;

static constexpr int Bc = 256, Sc = 512, Hc = 512, Pc = 96, G4c = 2048;

__device__ __forceinline__ v8f wmma_f32(v2f a, v2f b, v8f c) {
  // D = A(16x4 f32) * B(4x16 f32) + C(16x16 f32)
  return __builtin_amdgcn_wmma_f32_16x16x4_f32(false, a, false, b, (short)0, c,
                                               false, false);
}

// CDNA5 async global->LDS copy (16B per lane), tracked with ASYNCcnt.
// lds_addr: raw LDS byte address (low 32 bits of the generic pointer).
__device__ __forceinline__ void async_ld_b128(unsigned lds_addr,
                                              const float* gaddr) {
  asm volatile("global_load_async_to_lds_b128 %0, %1, off"
               :: "v"(lds_addr), "v"(gaddr) : "memory");
}
__device__ __forceinline__ void wait_async_le4() {
  asm volatile("s_wait_asynccnt 4" ::: "memory");
}
__device__ __forceinline__ void wait_async_0() {
  asm volatile("s_wait_asynccnt 0" ::: "memory");
}

__device__ __forceinline__ float sigm_f(float x) {
  return 1.f / (1.f + __expf(-x));
}
__device__ __forceinline__ float tanh_f(float x) {
  float e = __expf(-2.f * fabsf(x));
  float r = (1.f - e) / (1.f + e);
  return x < 0.f ? -r : r;
}

// --------------------------------------------------------------------------
// Generic [R][C] -> [C][R] fp32 transpose (tiled through LDS).
// --------------------------------------------------------------------------
__global__ __launch_bounds__(256) void transpose_kernel(
    const float* __restrict__ in, float* __restrict__ out, int R, int C) {
  __shared__ float tile[32][33];
  const int bx = blockIdx.x * 32;  // col base in 'in'
  const int by = blockIdx.y * 32;  // row base in 'in'
  const int tx = threadIdx.x, ty = threadIdx.y;  // 32 x 8
#pragma unroll
  for (int i = 0; i < 32; i += 8) {
    int r = by + ty + i, c = bx + tx;
    tile[ty + i][tx] = (r < R && c < C) ? in[(size_t)r * C + c] : 0.f;
  }
  __syncthreads();
#pragma unroll
  for (int i = 0; i < 32; i += 8) {
    int r = bx + ty + i, c = by + tx;  // out coords
    if (r < C && c < R) out[(size_t)r * R + c] = tile[tx][ty + i];
  }
}

__global__ void zero_kernel(float* __restrict__ p, size_t n) {
  size_t i = (size_t)blockIdx.x * blockDim.x + threadIdx.x;
  size_t stride = (size_t)gridDim.x * blockDim.x;
  for (; i < n; i += stride) p[i] = 0.f;
}

// --------------------------------------------------------------------------
// XP[M x 2048] = Xin[M x 512] @ WT[512 x 2048] + (bih[n] + bhh[n])
// 256 threads = 8 waves; block tile M=32, N=256; wave tile 16x64 (4 WMMA accs).
// A tile double-buffered in LDS via async b128 loads; row stride 132 dwords
// (==4 mod 64) so the 32-lane b64 A-frag read hits 64 distinct banks.
// --------------------------------------------------------------------------
__global__ __launch_bounds__(256) void xp_gemm_kernel(
    const float* __restrict__ Xin, const float* __restrict__ WT,
    const float* __restrict__ bih, const float* __restrict__ bhh,
    float* __restrict__ XP) {
  constexpr int K = Hc, N = G4c, KC = 128, RS = 132, NB = 32 * RS;
  __shared__ float sx[2 * NB];
  const unsigned sbase = (unsigned)(size_t)sx;  // raw LDS byte address
  const int m0 = blockIdx.x * 32;
  const int n0 = blockIdx.y * 256;
  const int lane = threadIdx.x & 31, wv = threadIdx.x >> 5;
  const int half = lane >> 4, lp = lane & 15;
  const int wm = wv >> 2;  // 0..1
  const int wn = wv & 3;   // 0..3
  const int mb = m0 + wm * 16;
  const int nb = n0 + wn * 64 + lp;

  const v8f vz = {0.f, 0.f, 0.f, 0.f, 0.f, 0.f, 0.f, 0.f};
  v8f acc[4] = {vz, vz, vz, vz};

  // issue one K-chunk (32 rows x 128 cols = 1024 float4; 4 asyncs per thread)
  auto issue = [&](int kc, int buf) {
#pragma unroll
    for (int i = 0; i < 4; ++i) {
      int f4 = threadIdx.x + i * 256;
      int row = f4 >> 5, c4 = f4 & 31;
      async_ld_b128(sbase + 4u * (unsigned)(buf * NB + row * RS + c4 * 4),
                    Xin + (size_t)(m0 + row) * K + kc + c4 * 4);
    }
  };

  issue(0, 0);
  for (int c = 0; c < K / KC; ++c) {
    if (c + 1 < K / KC) {
      issue((c + 1) * KC, (c + 1) & 1);
      wait_async_le4();  // in-order completion: current chunk is in LDS
    } else {
      wait_async_0();
    }
    __syncthreads();
    const float* sb = sx + (c & 1) * NB;
    const int kc = c * KC;
    for (int kk = 0; kk < KC; kk += 4) {
      v2f a = *(const v2f*)(&sb[(wm * 16 + lp) * RS + kk + 2 * half]);
      const float* wr = WT + (size_t)(kc + kk + 2 * half) * N + nb;
#pragma unroll
      for (int j = 0; j < 4; ++j) {
        v2f b;
        b.x = wr[j * 16];
        b.y = wr[N + j * 16];
        acc[j] = wmma_f32(a, b, acc[j]);
      }
    }
    __syncthreads();
  }
#pragma unroll
  for (int j = 0; j < 4; ++j) {
    const int n = nb + j * 16;
    const float bv = bih[n] + bhh[n];
#pragma unroll
    for (int r = 0; r < 8; ++r) {
      const int m = mb + r + 8 * half;
      XP[(size_t)m * N + n] = acc[j][r] + bv;
    }
  }
}

// --------------------------------------------------------------------------
// One LSTM timestep: gates[256x2048] = h_{t-1} @ WhhT, + xp (or layer-0 inline
// x*Wih0+bias), then fused cell update. 128 threads = 4 waves; block tile
// M=16 x 64 hidden cols; each wave holds the i/f/g/o 16x16 tiles for the SAME
// (row, hidden-col) block so the cell update is fully in-register.
// hprev row address = (m*hstride_s + t_prev)*H  (hstride_s=1 for flat [B,H]).
// --------------------------------------------------------------------------
__global__ __launch_bounds__(128) void lstm_step_kernel(
    const float* __restrict__ hprev, int hstride_s, int t_prev,
    const float* __restrict__ cprev, float* __restrict__ cnext,
    const float* __restrict__ WhhT,
    const float* __restrict__ xp,   // mode 1
    const float* __restrict__ x,    // mode 0: [B,S]
    const float* __restrict__ wih0, const float* __restrict__ bih0,
    const float* __restrict__ bhh0, float* __restrict__ outp, int t,
    int mode) {
  constexpr int K = Hc, N = G4c, KC = 128, RS = 132, NB = 16 * RS;
  __shared__ float sh[2 * NB];
  const unsigned sbase = (unsigned)(size_t)sh;
  const int m0 = blockIdx.x * 16;
  const int lane = threadIdx.x & 31, wv = threadIdx.x >> 5;
  const int half = lane >> 4, lp = lane & 15;
  const int hc = blockIdx.y * 64 + wv * 16 + lp;  // hidden column

  const v8f vz = {0.f, 0.f, 0.f, 0.f, 0.f, 0.f, 0.f, 0.f};
  v8f acc[4] = {vz, vz, vz, vz};  // i, f, g, o

  // issue one K-chunk (16 rows x 128 cols = 512 float4; 4 asyncs per thread)
  auto issue = [&](int kc, int buf) {
#pragma unroll
    for (int i = 0; i < 4; ++i) {
      int f4 = threadIdx.x + i * 128;
      int row = f4 >> 5, c4 = f4 & 31;
      async_ld_b128(
          sbase + 4u * (unsigned)(buf * NB + row * RS + c4 * 4),
          hprev + ((size_t)(m0 + row) * hstride_s + t_prev) * Hc + kc + c4 * 4);
    }
  };

  issue(0, 0);
  for (int c = 0; c < K / KC; ++c) {
    if (c + 1 < K / KC) {
      issue((c + 1) * KC, (c + 1) & 1);
      wait_async_le4();
    } else {
      wait_async_0();
    }
    __syncthreads();
    const float* sb = sh + (c & 1) * NB;
    const int kc = c * KC;
    for (int kk = 0; kk < KC; kk += 4) {
      v2f a = *(const v2f*)(&sb[lp * RS + kk + 2 * half]);
      const float* wr = WhhT + (size_t)(kc + kk + 2 * half) * N + hc;
#pragma unroll
      for (int gi = 0; gi < 4; ++gi) {
        v2f b;
        b.x = wr[gi * Hc];
        b.y = wr[N + gi * Hc];
        acc[gi] = wmma_f32(a, b, acc[gi]);
      }
    }
    __syncthreads();
  }

  float w0[4], bs[4];
  if (mode == 0) {
#pragma unroll
    for (int gi = 0; gi < 4; ++gi) {
      int cg = gi * Hc + hc;
      w0[gi] = wih0[cg];
      bs[gi] = bih0[cg] + bhh0[cg];
    }
  }
#pragma unroll
  for (int r = 0; r < 8; ++r) {
    const int m = m0 + r + 8 * half;
    float iv = acc[0][r], fv = acc[1][r], gv = acc[2][r], ov = acc[3][r];
    if (mode == 0) {
      float xv = x[(size_t)m * Sc + t];
      iv += xv * w0[0] + bs[0];
      fv += xv * w0[1] + bs[1];
      gv += xv * w0[2] + bs[2];
      ov += xv * w0[3] + bs[3];
    } else {
      const float* xr = xp + ((size_t)m * Sc + t) * N + hc;
      iv += xr[0];
      fv += xr[Hc];
      gv += xr[2 * Hc];
      ov += xr[3 * Hc];
    }
    float cp = cprev[(size_t)m * Hc + hc];
    float ig = sigm_f(iv), fg = sigm_f(fv), gg = tanh_f(gv), og = sigm_f(ov);
    float cn = fg * cp + ig * gg;
    float hn = og * tanh_f(cn);
    cnext[(size_t)m * Hc + hc] = cn;
    outp[((size_t)m * Sc + t) * Hc + hc] = hn;
  }
}

// --------------------------------------------------------------------------
// out[b][p] = h2[b, S-1, :] . Wfc[p, :] + bfc[p]   (256 x 96, tiny)
// --------------------------------------------------------------------------
__global__ __launch_bounds__(96) void fc_kernel(const float* __restrict__ h2,
                                                const float* __restrict__ Wfc,
                                                const float* __restrict__ bfc,
                                                float* __restrict__ out) {
  const int b = blockIdx.x, p = threadIdx.x;
  const float* hr = h2 + ((size_t)b * Sc + (Sc - 1)) * Hc;
  const float* wr = Wfc + (size_t)p * Hc;
  float acc = bfc[p];
  for (int k = 0; k < Hc; k += 4) {
    float4 h4 = *(const float4*)(hr + k);
    float4 w4 = *(const float4*)(wr + k);
    acc += h4.x * w4.x + h4.y * w4.y + h4.z * w4.z + h4.w * w4.w;
  }
  out[(size_t)b * Pc + p] = acc;
}

// --------------------------------------------------------------------------

extern "C" void kernel_launch(void* const* d_in, const int* in_sizes, int n_in,
                              void* d_out, int out_size, void* d_ws,
                              size_t ws_size, hipStream_t stream) {
  (void)in_sizes; (void)n_in; (void)out_size; (void)ws_size;
  const float* x    = (const float*)d_in[0];
  const float* Wih0 = (const float*)d_in[1];
  const float* Whh0 = (const float*)d_in[2];
  const float* bih0 = (const float*)d_in[3];
  const float* bhh0 = (const float*)d_in[4];
  const float* Wih1 = (const float*)d_in[5];
  const float* Whh1 = (const float*)d_in[6];
  const float* bih1 = (const float*)d_in[7];
  const float* bhh1 = (const float*)d_in[8];
  const float* Wih2 = (const float*)d_in[9];
  const float* Whh2 = (const float*)d_in[10];
  const float* bih2 = (const float*)d_in[11];
  const float* bhh2 = (const float*)d_in[12];
  const float* Wfc  = (const float*)d_in[13];
  const float* bfc  = (const float*)d_in[14];

  float* w = (float*)d_ws;
  const size_t WT_SZ = (size_t)Hc * G4c;   // 1,048,576
  const size_t BH    = (size_t)Bc * Hc;    // 131,072
  const size_t SEQ   = (size_t)Bc * Sc * Hc;      // 67,108,864
  float* WhhT0 = w; w += WT_SZ;
  float* WhhT1 = w; w += WT_SZ;
  float* WhhT2 = w; w += WT_SZ;
  float* WihT1 = w; w += WT_SZ;
  float* WihT2 = w; w += WT_SZ;
  float* cbuf  = w; w += BH;
  float* c0buf = w; w += BH;
  float* zbuf  = w; w += BH;
  float* out0  = w; w += SEQ;
  float* out1  = w; w += SEQ;
  float* xp    = w; w += (size_t)Bc * Sc * G4c;

  // Pre-transpose weights: [2048][512] -> [512][2048] (coalesced B-frags).
  dim3 tb(32, 8), tg(Hc / 32, G4c / 32);  // (16, 64)
  transpose_kernel<<<tg, tb, 0, stream>>>(Whh0, WhhT0, G4c, Hc);
  transpose_kernel<<<tg, tb, 0, stream>>>(Whh1, WhhT1, G4c, Hc);
  transpose_kernel<<<tg, tb, 0, stream>>>(Whh2, WhhT2, G4c, Hc);
  transpose_kernel<<<tg, tb, 0, stream>>>(Wih1, WihT1, G4c, Hc);
  transpose_kernel<<<tg, tb, 0, stream>>>(Wih2, WihT2, G4c, Hc);
  zero_kernel<<<64, 256, 0, stream>>>(zbuf, BH);

  const dim3 sg(Bc / 16, Hc / 64);         // (16, 8), 128 blocks/step
  const dim3 xg(Bc * Sc / 32, G4c / 256);  // (4096, 8)

  // ---- layer 0 (zero init state; input projection inlined since in=1) ----
  for (int t = 0; t < Sc; ++t) {
    const float* hp = t ? out0 : zbuf;
    const int hs = t ? Sc : 1;
    const int tp = t ? t - 1 : 0;
    const float* cp = t ? cbuf : zbuf;
    lstm_step_kernel<<<sg, 128, 0, stream>>>(hp, hs, tp, cp, cbuf, WhhT0,
                                             nullptr, x, Wih0, bih0, bhh0,
                                             out0, t, 0);
  }
  // snapshot layer-0 final cell state (reused as init for layers 1 and 2)
  hipMemcpyAsync(c0buf, cbuf, BH * sizeof(float), hipMemcpyDeviceToDevice,
                 stream);

  // ---- layer 1: xp1 = out0 @ Wih1^T + bias, then scan ----
  xp_gemm_kernel<<<xg, 256, 0, stream>>>(out0, WihT1, bih1, bhh1, xp);
  for (int t = 0; t < Sc; ++t) {
    const float* hp = t ? out1 : out0;  // t=0: layer-0 final h slice
    const int tp = t ? t - 1 : Sc - 1;
    const float* cp = t ? cbuf : c0buf;
    lstm_step_kernel<<<sg, 128, 0, stream>>>(hp, Sc, tp, cp, cbuf, WhhT1, xp,
                                             nullptr, nullptr, nullptr,
                                             nullptr, out1, t, 1);
  }

  // ---- layer 2: xp2 = out1 @ Wih2^T + bias; output reuses out0 buffer ----
  xp_gemm_kernel<<<xg, 256, 0, stream>>>(out1, WihT2, bih2, bhh2, xp);
  for (int t = 0; t < Sc; ++t) {
    // t=0 reads out0[:,S-1,:] (layer-0 final h); t>0 reads own prior slice.
    const int tp = t ? t - 1 : Sc - 1;
    const float* cp = t ? cbuf : c0buf;
    lstm_step_kernel<<<sg, 128, 0, stream>>>(out0, Sc, tp, cp, cbuf, WhhT2, xp,
                                             nullptr, nullptr, nullptr,
                                             nullptr, out0, t, 1);
  }

  // ---- FC head ----
  fc_kernel<<<Bc, Pc, 0, stream>>>(out0, Wfc, bfc, (float*)d_out);
}